// DenseCaps_v2_8151847927909
// MI455X (gfx1250) — compile-verified
//
#include <hip/hip_runtime.h>

// DenseCaps dynamic routing, fully fused per routing iteration.
// u_hat ([256,2048,160] f32 = 335 MB) is NEVER materialized: it is recomputed
// in WMMA accumulators each phase, so the working set (x 16.8MB + W 10.5MB +
// bB 21MB) stays resident in the 192 MB L2 instead of sweeping 2.3 GB of HBM.

#define NCAP 10
#define RR   2048
#define IND  8
#define OUTD 16
#define BATCH 256
#define CEPS 1e-8f

typedef __attribute__((ext_vector_type(2))) float v2f;
typedef __attribute__((ext_vector_type(8))) float v8f;

// Wave-local LDS ordering: DS ops of one wave complete in order; a dscnt wait
// plus a compiler memory barrier is enough for "lane A stored, lane B loads"
// within the same wave (agrS/cS are per-wave-private regions).
__device__ __forceinline__ void wave_lds_fence() {
    asm volatile("s_wait_dscnt 0x0" ::: "memory");
}

// ---------------------------------------------------------------------------
// zero s accumulator
__global__ void dc_zero_kernel(float* __restrict__ p, int n) {
    int i = blockIdx.x * blockDim.x + threadIdx.x;
    if (i < n) p[i] = 0.0f;
}

// ---------------------------------------------------------------------------
// Fused routing iteration.
//  PHASE 0: c = softmax(b_in[r]);                 s += c * u_hat
//  PHASE 1: bB = b_in[r] + <u_hat,v>; write bB;   c = softmax(bB); s += c*u_hat
//  PHASE 2: bB = bB_ws   + <u_hat,v>; write bB;   c = softmax(bB); s += c*u_hat
//  PHASE 3: bB = bB_ws   + <u_hat,v>; (no write); c = softmax(bB) -> d_out
//
// Block = 128 threads = 4 waves. blockIdx.y picks a 16-batch tile,
// blockIdx.x a 32-route chunk; each wave owns 8 routes. Per route, one wave
// computes the 16x160 u_hat tile with 20 V_WMMA_F32_16X16X4_F32 ops
// (10 capsule tiles x 2 K-steps of 4).
template <int PHASE>
__global__ void __launch_bounds__(128)
dc_route_iter(const float* __restrict__ x, const float* __restrict__ W,
              const float* __restrict__ b_in, float* __restrict__ bB,
              const float* __restrict__ v_ws, float* __restrict__ s_ws,
              float* __restrict__ c_out)
{
    __shared__ __align__(16) float vS[16 * 160];  // v slice for this batch tile
    __shared__ float agrS[4][16][NCAP];           // per-wave agreement scratch
    __shared__ float cS[4][16][NCAP];             // per-wave coupling coeffs

    const int wave  = threadIdx.x >> 5;
    const int lane  = threadIdx.x & 31;
    const int half  = lane >> 4;          // 0 or 1 (16-lane half)
    const int l16   = lane & 15;
    const int btile = blockIdx.y * 16;
    const int rbase = blockIdx.x * 32 + wave * 8;

    if (PHASE >= 1) {
        // Stage v[btile..btile+15][0..159] (10 KB, rows contiguous) into LDS
        // with the CDNA5 async copy path (tracked by ASYNCcnt, not LOADcnt).
        const float* gsrc = v_ws + (size_t)btile * 160;
        for (int i = threadIdx.x; i < 640; i += 128) {   // 640 x 16B = 10 KB
            // Generic pointers to LDS carry the LDS byte offset in addr[31:0].
            unsigned int ldsoff = (unsigned int)(size_t)&vS[4 * i];
            unsigned long long gaddr = (unsigned long long)(size_t)(gsrc + 4 * i);
            asm volatile("global_load_async_to_lds_b128 %0, %1, off"
                         :: "v"(ldsoff), "v"(gaddr) : "memory");
        }
        asm volatile("s_wait_asynccnt 0x0" ::: "memory");
    }
    __syncthreads();   // only block-wide barrier: vS is read-only afterwards

    v8f s_acc[NCAP];
    #pragma unroll
    for (int n = 0; n < NCAP; ++n) s_acc[n] = {};

    for (int rr = 0; rr < 8; ++rr) {
        const int r = rbase + rr;

        // ---- A operand: x[btile+M, r, 0:8].  16x4 f32 layout:
        // lanes 0-15: M=0..15, VGPR0/1 = K0/K1; lanes 16-31: K2/K3.
        const float* xp = x + ((size_t)(btile + l16) * RR + r) * IND;
        const int k0 = half * 2;
        const v2f a0 = *(const v2f*)(xp + k0);        // K-step 0..3
        const v2f a1 = *(const v2f*)(xp + k0 + 4);    // K-step 4..7

        const float* Wp = W + (size_t)r * (IND * NCAP * OUTD);
        if (rr + 1 < 8)   // pull next route's weights toward the WGP
            __builtin_prefetch(W + (size_t)(r + 1) * (IND * NCAP * OUTD), 0, 3);

        // ---- u_hat tile: C[n] holds [16 batches x 16 od] for capsule n.
        v8f C[NCAP];
        #pragma unroll
        for (int n = 0; n < NCAP; ++n) {
            const int col = n * 16 + l16;     // N column (od)
            v2f b0; b0.x = Wp[(k0 + 0) * 160 + col]; b0.y = Wp[(k0 + 1) * 160 + col];
            v2f b1; b1.x = Wp[(k0 + 4) * 160 + col]; b1.y = Wp[(k0 + 5) * 160 + col];
            v8f acc = {};
            acc = __builtin_amdgcn_wmma_f32_16x16x4_f32(
                    false, a0, false, b0, (short)0, acc, false, false);
            acc = __builtin_amdgcn_wmma_f32_16x16x4_f32(
                    false, a1, false, b1, (short)0, acc, false, false);
            C[n] = acc;
        }

        // ---- agreement: agr[b, nc] = sum_od u_hat * v  (16-lane butterfly)
        if (PHASE >= 1) {
            #pragma unroll
            for (int n = 0; n < NCAP; ++n) {
                #pragma unroll
                for (int j = 0; j < 8; ++j) {
                    const int m = j + 8 * half;           // batch row in tile
                    float p = C[n][j] * vS[m * 160 + n * 16 + l16];
                    p += __shfl_xor(p, 1);
                    p += __shfl_xor(p, 2);
                    p += __shfl_xor(p, 4);
                    p += __shfl_xor(p, 8);
                    if (l16 == n) agrS[wave][m][n] = p;
                }
            }
            wave_lds_fence();   // per-wave region: no block barrier needed
        }

        // ---- softmax over the 10 classes; lane m handles batch row m
        if (lane < 16) {
            const int m = lane;
            const size_t brn = ((size_t)(btile + m) * RR + r) * NCAP;
            float bv[NCAP];
            if (PHASE == 0 || PHASE == 1) {
                #pragma unroll
                for (int h = 0; h < 5; ++h) {     // b_in row, 5 x b64
                    const v2f t = ((const v2f*)(b_in + (size_t)r * NCAP))[h];
                    bv[2 * h] = t.x; bv[2 * h + 1] = t.y;
                }
            } else {
                #pragma unroll
                for (int h = 0; h < 5; ++h) {     // bB row, 5 x b64
                    const v2f t = ((const v2f*)(bB + brn))[h];
                    bv[2 * h] = t.x; bv[2 * h + 1] = t.y;
                }
            }
            if (PHASE >= 1) {
                #pragma unroll
                for (int n = 0; n < NCAP; ++n) bv[n] += agrS[wave][m][n];
            }
            if (PHASE == 1 || PHASE == 2) {
                #pragma unroll
                for (int h = 0; h < 5; ++h) {
                    v2f t; t.x = bv[2 * h]; t.y = bv[2 * h + 1];
                    ((v2f*)(bB + brn))[h] = t;
                }
            }
            float mx = bv[0];
            #pragma unroll
            for (int n = 1; n < NCAP; ++n) mx = fmaxf(mx, bv[n]);
            float e[NCAP], sum = 0.0f;
            #pragma unroll
            for (int n = 0; n < NCAP; ++n) { e[n] = __expf(bv[n] - mx); sum += e[n]; }
            const float inv = 1.0f / sum;
            float cv[NCAP];
            #pragma unroll
            for (int n = 0; n < NCAP; ++n) {
                cv[n] = e[n] * inv;
                cS[wave][m][n] = cv[n];
            }
            if (PHASE == 3) {
                #pragma unroll
                for (int h = 0; h < 5; ++h) {
                    v2f t; t.x = cv[2 * h]; t.y = cv[2 * h + 1];
                    ((v2f*)(c_out + brn))[h] = t;
                }
            }
        }
        wave_lds_fence();

        // ---- s += c * u_hat, in the same C-register layout
        #pragma unroll
        for (int n = 0; n < NCAP; ++n) {
            #pragma unroll
            for (int j = 0; j < 8; ++j) {
                const int m = j + 8 * half;
                s_acc[n][j] += cS[wave][m][n] * C[n][j];
            }
        }
        // next iteration's DS writes are same-wave in-order; fence at the top
        // of the softmax section covers the reuse of agrS/cS.
    }

    // ---- flush s partials: one f32 atomic per distinct (b, nc, od) element
    #pragma unroll
    for (int n = 0; n < NCAP; ++n) {
        #pragma unroll
        for (int j = 0; j < 8; ++j) {
            const int b = btile + j + 8 * half;
            atomicAdd(&s_ws[((size_t)b * NCAP + n) * OUTD + l16], s_acc[n][j]);
        }
    }
}

// ---------------------------------------------------------------------------
// v = squash(s); resets s to 0 for the next accumulation phase.
// 160 threads per block = 10 groups of 16 lanes (aligned to half-waves).
template <bool FINAL>
__global__ void __launch_bounds__(160)
dc_squash(float* __restrict__ s_ws, float* __restrict__ v_dst)
{
    const int b = blockIdx.x;
    const int t = threadIdx.x;           // nc*16 + od
    const size_t idx = (size_t)b * 160 + t;
    float val = s_ws[idx];
    float sq = val * val;
    sq += __shfl_xor(sq, 1);
    sq += __shfl_xor(sq, 2);
    sq += __shfl_xor(sq, 4);
    sq += __shfl_xor(sq, 8);             // ||s||^2 over od
    const float norm = sqrtf(sq);
    v_dst[idx] = norm / (1.0f + sq + CEPS) * val;
    s_ws[idx] = 0.0f;
}

// ---------------------------------------------------------------------------
extern "C" void kernel_launch(void* const* d_in, const int* in_sizes, int n_in,
                              void* d_out, int out_size, void* d_ws, size_t ws_size,
                              hipStream_t stream)
{
    (void)in_sizes; (void)n_in; (void)out_size; (void)ws_size;
    const float* x    = (const float*)d_in[0];   // [256, 2048, 8]
    const float* W    = (const float*)d_in[1];   // [2048, 8, 160]
    const float* b_in = (const float*)d_in[2];   // [2048, 10]

    float* out   = (float*)d_out;
    float* v_out = out;                          // [256, 10, 16]
    float* c_out = out + (size_t)BATCH * NCAP * OUTD;  // [256, 2048, 10]

    float* ws   = (float*)d_ws;
    float* s_ws = ws;                            // 40960 f32
    float* v_ws = ws + 40960;                    // 40960 f32
    float* bB   = ws + 81920;                    // 256*2048*10 f32 (21 MB)

    const dim3 grid(RR / 32, BATCH / 16);        // (64, 16)
    const dim3 blk(128);

    dc_zero_kernel<<<(BATCH * NCAP * OUTD + 255) / 256, 256, 0, stream>>>(
        s_ws, BATCH * NCAP * OUTD);

    dc_route_iter<0><<<grid, blk, 0, stream>>>(x, W, b_in, bB, v_ws, s_ws, c_out);
    dc_squash<false><<<BATCH, 160, 0, stream>>>(s_ws, v_ws);

    dc_route_iter<1><<<grid, blk, 0, stream>>>(x, W, b_in, bB, v_ws, s_ws, c_out);
    dc_squash<false><<<BATCH, 160, 0, stream>>>(s_ws, v_ws);

    dc_route_iter<2><<<grid, blk, 0, stream>>>(x, W, b_in, bB, v_ws, s_ws, c_out);
    dc_squash<false><<<BATCH, 160, 0, stream>>>(s_ws, v_ws);

    dc_route_iter<3><<<grid, blk, 0, stream>>>(x, W, b_in, bB, v_ws, s_ws, c_out);
    dc_squash<true><<<BATCH, 160, 0, stream>>>(s_ws, v_out);
}